// GraphAttentionLayer_64939905515824
// MI455X (gfx1250) — compile-verified
//
#include <hip/hip_runtime.h>

#define N_NODES   100000
#define N_EDGES   1600000
#define E_TOTAL   (N_EDGES + N_NODES)   // edges + self loops
#define F_IN      128
#define HEADS     4
#define C_OUT     32
#define HC        (HEADS * C_OUT)       // 128
#define NEG_SLOPE 0.2f
#define M_TILES   (N_NODES / 16)        // 6250 exactly

typedef __attribute__((ext_vector_type(2))) float v2f;
typedef __attribute__((ext_vector_type(8))) float v8f;

// ---------------------------------------------------------------------------
// Kernel 1: h = x @ W via V_WMMA_F32_16X16X4_F32 (exact f32 matrix pipe).
// One wave -> 16 rows of x, all 128 output cols (8 N-tiles), K looped by 4.
// W (64KB) staged in LDS; A frags loaded per the 32-bit A-matrix 16x4 layout.
// ---------------------------------------------------------------------------
__global__ __launch_bounds__(256) void gat_gemm(const float* __restrict__ x,
                                                const float* __restrict__ W,
                                                float* __restrict__ h) {
    __shared__ float Wlds[F_IN * HC];   // 64 KB of the 320 KB WGP LDS
    {
        const float4* Wv = (const float4*)W;
        float4*       Lv = (float4*)Wlds;
        for (int i = threadIdx.x; i < (F_IN * HC) / 4; i += 256) Lv[i] = Wv[i];
    }
    __syncthreads();

    const int wave  = threadIdx.x >> 5;
    const int lane  = threadIdx.x & 31;
    const int half  = lane >> 4;        // 0: K pair {0,1}; 1: K pair {2,3}
    const int r     = lane & 15;        // row within tile / col within N-tile
    const int mtile = blockIdx.x * 8 + wave;
    if (mtile >= M_TILES) return;

    const float* xrow = x + (size_t)(mtile * 16 + r) * F_IN;
    __builtin_prefetch(xrow, 0, 0);     // global_prefetch_b8

    v8f acc[8];
    const v8f zero = {0.f, 0.f, 0.f, 0.f, 0.f, 0.f, 0.f, 0.f};
#pragma unroll
    for (int nt = 0; nt < 8; ++nt) acc[nt] = zero;

    for (int k = 0; k < F_IN; k += 4) {
        const int kk = k + half * 2;
        v2f a;
        a.x = xrow[kk];
        a.y = xrow[kk + 1];
#pragma unroll
        for (int nt = 0; nt < 8; ++nt) {
            const int col = nt * 16 + r;
            v2f b;
            b.x = Wlds[kk * HC + col];
            b.y = Wlds[(kk + 1) * HC + col];
            acc[nt] = __builtin_amdgcn_wmma_f32_16x16x4_f32(
                false, a, false, b, (short)0, acc[nt], false, false);
        }
    }

    // C/D layout: lanes 0-15 -> M = v, N = lane; lanes 16-31 -> M = 8+v, N = lane-16
#pragma unroll
    for (int nt = 0; nt < 8; ++nt) {
        float* outp = h + (size_t)(mtile * 16 + half * 8) * HC + nt * 16 + r;
#pragma unroll
        for (int v = 0; v < 8; ++v) outp[(size_t)v * HC] = acc[nt][v];
    }
}

// ---------------------------------------------------------------------------
// Kernel 2: a_src[n,h] = h[n,h,:] . att_src[h,:]   (and same for dst)
// ---------------------------------------------------------------------------
__global__ __launch_bounds__(256) void gat_attn(const float* __restrict__ h,
                                                const float* __restrict__ att_src,
                                                const float* __restrict__ att_dst,
                                                float* __restrict__ asrc,
                                                float* __restrict__ adst) {
    int t = blockIdx.x * 256 + threadIdx.x;     // one (node, head) per thread
    if (t >= N_NODES * HEADS) return;
    const int n  = t >> 2;
    const int hh = t & 3;
    const float4* hp = (const float4*)(h + (size_t)n * HC + hh * C_OUT);
    const float4* as = (const float4*)(att_src + hh * C_OUT);
    const float4* ad = (const float4*)(att_dst + hh * C_OUT);
    float ss = 0.f, sd = 0.f;
#pragma unroll
    for (int i = 0; i < C_OUT / 4; ++i) {
        float4 hv = hp[i], a = as[i], d = ad[i];
        ss += hv.x * a.x + hv.y * a.y + hv.z * a.z + hv.w * a.w;
        sd += hv.x * d.x + hv.y * d.y + hv.z * d.z + hv.w * d.w;
    }
    asrc[t] = ss;
    adst[t] = sd;
}

// ---------------------------------------------------------------------------
// Kernel 3: init out=0, m=-inf, s=0
// ---------------------------------------------------------------------------
__global__ __launch_bounds__(256) void gat_init(float* __restrict__ out,
                                                float* __restrict__ m,
                                                float* __restrict__ s) {
    int i = blockIdx.x * 256 + threadIdx.x;
    if (i < N_NODES * HC) out[i] = 0.f;
    if (i < N_NODES * HEADS) { m[i] = -__builtin_inff(); s[i] = 0.f; }
}

__device__ __forceinline__ void atomicMaxF32(float* addr, float v) {
    // sign-split trick: correct total order on IEEE bits, init -inf safe
    if (v >= 0.f) atomicMax((int*)addr, __float_as_int(v));
    else          atomicMin((unsigned int*)addr, __float_as_uint(v));
}

__device__ __forceinline__ void edge_endpoints(const int* __restrict__ ei,
                                               int e, int& src, int& dst) {
    if (e < N_EDGES) { src = ei[e]; dst = ei[N_EDGES + e]; }
    else             { src = dst = e - N_EDGES; }          // self loop
}

// ---------------------------------------------------------------------------
// Kernel 4: segment max of leaky_relu(a_src[src]+a_dst[dst]) over dst
// ---------------------------------------------------------------------------
__global__ __launch_bounds__(256) void gat_edge_max(const int* __restrict__ ei,
                                                    const float* __restrict__ asrc,
                                                    const float* __restrict__ adst,
                                                    float* __restrict__ m) {
    int e = blockIdx.x * 256 + threadIdx.x;
    if (e >= E_TOTAL) return;
    int src, dst;
    edge_endpoints(ei, e, src, dst);
#pragma unroll
    for (int hh = 0; hh < HEADS; ++hh) {
        float v = asrc[src * HEADS + hh] + adst[dst * HEADS + hh];
        v = v > 0.f ? v : NEG_SLOPE * v;
        atomicMaxF32(&m[dst * HEADS + hh], v);
    }
}

// ---------------------------------------------------------------------------
// Kernel 5: segment sum of exp(e - m[dst]) over dst
// ---------------------------------------------------------------------------
__global__ __launch_bounds__(256) void gat_edge_sum(const int* __restrict__ ei,
                                                    const float* __restrict__ asrc,
                                                    const float* __restrict__ adst,
                                                    const float* __restrict__ m,
                                                    float* __restrict__ s) {
    int e = blockIdx.x * 256 + threadIdx.x;
    if (e >= E_TOTAL) return;
    int src, dst;
    edge_endpoints(ei, e, src, dst);
#pragma unroll
    for (int hh = 0; hh < HEADS; ++hh) {
        float v = asrc[src * HEADS + hh] + adst[dst * HEADS + hh];
        v = v > 0.f ? v : NEG_SLOPE * v;
        atomicAdd(&s[dst * HEADS + hh], __expf(v - m[dst * HEADS + hh]));
    }
}

// ---------------------------------------------------------------------------
// Kernel 6: message scatter: out[dst] += h[src] * alpha  (128 thr/edge)
// ---------------------------------------------------------------------------
__global__ __launch_bounds__(256) void gat_scatter(const int* __restrict__ ei,
                                                   const float* __restrict__ h,
                                                   const float* __restrict__ asrc,
                                                   const float* __restrict__ adst,
                                                   const float* __restrict__ m,
                                                   const float* __restrict__ s,
                                                   float* __restrict__ out) {
    int e = blockIdx.x * 2 + (threadIdx.x >> 7);
    int c = threadIdx.x & 127;
    if (e >= E_TOTAL) return;
    int src, dst;
    edge_endpoints(ei, e, src, dst);
    const int hh = c >> 5;   // head of this channel
    float v = asrc[src * HEADS + hh] + adst[dst * HEADS + hh];
    v = v > 0.f ? v : NEG_SLOPE * v;
    const float alpha = __expf(v - m[dst * HEADS + hh]) /
                        (s[dst * HEADS + hh] + 1e-16f);
    atomicAdd(&out[(size_t)dst * HC + c], h[(size_t)src * HC + c] * alpha);
}

// ---------------------------------------------------------------------------
// Kernel 7: out = tanh(out + bias)
// ---------------------------------------------------------------------------
__global__ __launch_bounds__(256) void gat_final(float* __restrict__ out,
                                                 const float* __restrict__ bias) {
    int i = blockIdx.x * 256 + threadIdx.x;
    if (i >= N_NODES * HC) return;
    out[i] = tanhf(out[i] + bias[i & (HC - 1)]);
}

// ---------------------------------------------------------------------------
extern "C" void kernel_launch(void* const* d_in, const int* in_sizes, int n_in,
                              void* d_out, int out_size, void* d_ws, size_t ws_size,
                              hipStream_t stream) {
    const float* x       = (const float*)d_in[0];
    const int*   ei      = (const int*)d_in[1];
    const float* W       = (const float*)d_in[2];
    const float* att_src = (const float*)d_in[3];
    const float* att_dst = (const float*)d_in[4];
    const float* bias    = (const float*)d_in[5];
    float*       out     = (float*)d_out;

    // workspace carve-up
    char* ws = (char*)d_ws;
    float* h    = (float*)ws;                                  ws += (size_t)N_NODES * HC * 4;
    float* asrc = (float*)ws;                                  ws += (size_t)N_NODES * HEADS * 4;
    float* adst = (float*)ws;                                  ws += (size_t)N_NODES * HEADS * 4;
    float* m    = (float*)ws;                                  ws += (size_t)N_NODES * HEADS * 4;
    float* s    = (float*)ws;

    // 1) h = x @ W  (WMMA f32)
    hipLaunchKernelGGL(gat_gemm, dim3((M_TILES + 7) / 8), dim3(256), 0, stream, x, W, h);
    // 2) attention logits
    hipLaunchKernelGGL(gat_attn, dim3((N_NODES * HEADS + 255) / 256), dim3(256), 0, stream,
                       h, att_src, att_dst, asrc, adst);
    // 3) init accumulators
    hipLaunchKernelGGL(gat_init, dim3((N_NODES * HC + 255) / 256), dim3(256), 0, stream,
                       out, m, s);
    // 4) segment max
    hipLaunchKernelGGL(gat_edge_max, dim3((E_TOTAL + 255) / 256), dim3(256), 0, stream,
                       ei, asrc, adst, m);
    // 5) segment sum of exp
    hipLaunchKernelGGL(gat_edge_sum, dim3((E_TOTAL + 255) / 256), dim3(256), 0, stream,
                       ei, asrc, adst, m, s);
    // 6) weighted message scatter
    hipLaunchKernelGGL(gat_scatter, dim3((E_TOTAL + 1) / 2), dim3(256), 0, stream,
                       ei, h, asrc, adst, m, s, out);
    // 7) bias + tanh
    hipLaunchKernelGGL(gat_final, dim3((N_NODES * HC + 255) / 256), dim3(256), 0, stream,
                       out, bias);
}